// GroupedLinearsAdvanced_81192061763930
// MI455X (gfx1250) — compile-verified
//
#include <hip/hip_runtime.h>

typedef __attribute__((ext_vector_type(2))) float v2f;
typedef __attribute__((ext_vector_type(8))) float v8f;

#define B_DIM  16
#define IN_DIM 128
#define OUT_DIM 128
#define D_TOT  4096
#define LDA    132   // padded LDS row stride (floats): 132*4 B, keeps 8B align, breaks 64-bank conflicts

__global__ __launch_bounds__(256)
void grouped_linears_wmma_f32(const float* __restrict__ x,
                              const float* __restrict__ W,
                              const float* __restrict__ bias,
                              float* __restrict__ out) {
    __shared__ float ldsA[B_DIM * LDA];   // A tile: x[:, :, d] as [16 x 128]

    const int d    = blockIdx.x;          // channel
    const int t    = threadIdx.x;
    const int wave = t >> 5;              // 0..7 -> out_d tile
    const int lane = t & 31;

    // ---- Stage A = x[:, :, d] into LDS: x[b, i, d] @ b*IN*D + i*D + d ----
    // 2048 elements, 8 per thread. Strided in HBM but L2-resident across blocks.
    {
        const size_t dbase = (size_t)d;
        #pragma unroll
        for (int j = 0; j < 8; ++j) {
            int e  = t + 256 * j;         // 0..2047
            int bb = e >> 7;              // batch row 0..15
            int ii = e & 127;             // in_d 0..127
            ldsA[bb * LDA + ii] =
                x[dbase + (size_t)bb * ((size_t)IN_DIM * D_TOT) + (size_t)ii * D_TOT];
        }
    }
    __syncthreads();

    // ---- Fragment geometry (wave32 WMMA f32 16x16x4) ----
    // A 16x4: lane<16 -> M=lane, K=k0+{0,1}; lane>=16 -> M=lane-16, K=k0+{2,3}
    // B 4x16: V0 = K=k0 (low lanes) / k0+2 (high); V1 = k0+1 / k0+3; N = lane%16
    // C/D 16x16: VGPR r -> M = r + 8*(lane>=16), N = lane%16
    const int o0  = wave * 16;
    const int n   = lane & 15;
    const int hi  = lane >> 4;            // 0: lanes 0-15, 1: lanes 16-31
    const int kd  = hi * 2;               // K offset within group-of-4

    const float* __restrict__ Wd = W + (size_t)d * ((size_t)IN_DIM * OUT_DIM);
    const float* __restrict__ Arow = &ldsA[n * LDA];

    v8f acc = {};
    #pragma unroll 8
    for (int k0 = 0; k0 < IN_DIM; k0 += 4) {
        v2f a;
        a.x = Arow[k0 + kd];
        a.y = Arow[k0 + kd + 1];
        v2f bf;
        bf.x = Wd[(size_t)(k0 + kd)     * OUT_DIM + o0 + n];
        bf.y = Wd[(size_t)(k0 + kd + 1) * OUT_DIM + o0 + n];
        // D = A * B + C   (f32, 16x16x4)
        acc = __builtin_amdgcn_wmma_f32_16x16x4_f32(
                  /*neg_a=*/false, a, /*neg_b=*/false, bf,
                  /*c_mod=*/(short)0, acc, /*reuse_a=*/false, /*reuse_b=*/false);
    }

    // ---- Epilogue: add bias, scatter to out[b, o, d] @ b*OUT*D + o*D + d ----
    const float bv = bias[(size_t)d * OUT_DIM + o0 + n];
    const int   m0 = hi * 8;
    const size_t ocol = (size_t)(o0 + n) * D_TOT + (size_t)d;
    #pragma unroll
    for (int r = 0; r < 8; ++r) {
        int m = m0 + r;
        out[(size_t)m * ((size_t)OUT_DIM * D_TOT) + ocol] = acc[r] + bv;
    }
}

extern "C" void kernel_launch(void* const* d_in, const int* in_sizes, int n_in,
                              void* d_out, int out_size, void* d_ws, size_t ws_size,
                              hipStream_t stream) {
    const float* x    = (const float*)d_in[0];
    const float* W    = (const float*)d_in[1];
    const float* bias = (const float*)d_in[2];
    float*       out  = (float*)d_out;

    grouped_linears_wmma_f32<<<D_TOT, 256, 0, stream>>>(x, W, bias, out);
}